// GraphAttentionLayer_75419625717976
// MI455X (gfx1250) — compile-verified
//
#include <hip/hip_runtime.h>
#include <math.h>

#define ALPHA 0.2f
#define EPSV  1e-6f
#define DIN   256
#define F     64
#define KC    64

typedef float v2f __attribute__((ext_vector_type(2)));
typedef float v8f __attribute__((ext_vector_type(8)));

__device__ __forceinline__ float leaky(float x) { return x >= 0.f ? x : ALPHA * x; }

// ---------------------------------------------------------------- zero
__global__ void k_zero(float* __restrict__ out, float* __restrict__ rsh,
                       float* __restrict__ rsl, int N) {
    long long i = (long long)blockIdx.x * blockDim.x + threadIdx.x;
    long long stride = (long long)gridDim.x * blockDim.x;
    long long nOut = (long long)N * 128;
    for (long long j = i; j < nOut; j += stride) out[j] = 0.f;
    for (long long j = i; j < N; j += stride) { rsh[j] = 0.f; rsl[j] = 0.f; }
}

// ------------------------------------------------- fold a-vectors, norms, theta
// params layout: [0..6F) = 6 folded coefficient vectors of length F
//  [6F+0]=1/||a_high||  [6F+1]=1/||a_low||  [6F+2]=theta_high  [6F+3]=theta_low
__global__ void k_params(const float* __restrict__ ah, const float* __restrict__ al,
                         const float* __restrict__ ch, const float* __restrict__ cl,
                         float* __restrict__ params) {
    __shared__ float red[256];
    int t = threadIdx.x;
    float vh = ah[t], vl = al[t];
    red[t] = vh * vh; __syncthreads();
    for (int s = 128; s > 0; s >>= 1) { if (t < s) red[t] += red[t + s]; __syncthreads(); }
    float nh = sqrtf(red[0]); __syncthreads();
    red[t] = vl * vl; __syncthreads();
    for (int s = 128; s > 0; s >>= 1) { if (t < s) red[t] += red[t + s]; __syncthreads(); }
    float nl = sqrtf(red[0]);
    if (t < F) {
        params[0*F + t] = ah[t]     + ah[2*F + t] + ah[3*F + t];   // high, src coeff
        params[1*F + t] = ah[F + t] + ah[2*F + t] - ah[3*F + t];   // high, dst coeff
        params[2*F + t] = al[t];                                   // low via h_high, src
        params[3*F + t] = al[F + t];                               // low via h_high, dst
        params[4*F + t] = al[2*F + t] + al[3*F + t];               // low via h_low, src
        params[5*F + t] = al[2*F + t] - al[3*F + t];               // low via h_low, dst
    }
    if (t == 0) {
        params[6*F + 0] = 1.f / nh;
        params[6*F + 1] = 1.f / nl;
        float c = ch[0];
        params[6*F + 2] = (fminf(fmaxf(c + 3.f, 0.f), 6.f) / 3.f + EPSV) * 0.5f;
        c = cl[0];
        params[6*F + 3] = (fminf(fmaxf(c + 3.f, 0.f), 6.f) / 3.f + EPSV) * 0.5f;
    }
}

// ------------------------------------------------- H = leaky(X @ W), WMMA fp32
// Block = 128 threads = 4 waves; each wave owns a 16-row x 64-col output tile.
// W tile staged in LDS in K-pair layout so each B fragment is ONE contiguous
// 8-byte ds load straight into an even-aligned VGPR pair (no repack movs).
__global__ void __launch_bounds__(128)
k_gemm_leaky(const float* __restrict__ X, const float* __restrict__ W,
             float* __restrict__ H, int N) {
    // sW2[k2][n] = (W[kc+2*k2][n], W[kc+2*k2+1][n]); row stride 80 float2
    // = 160 banks === 32 (mod 64): the two half-waves (adjacent k2 rows) hit
    // disjoint 32-bank windows -> conflict-free b64 access.
    __shared__ v2f   sW2[KC / 2][F + 16];
    __shared__ float sA[4][16][KC + 4];

    const int tid  = threadIdx.x;
    const int wave = tid >> 5;
    const int lane = tid & 31;
    const int half = lane >> 4;
    const int l16  = lane & 15;
    const int tileRow = (blockIdx.x * 4 + wave) * 16;

    v8f acc0 = {}, acc1 = {}, acc2 = {}, acc3 = {};

    for (int kc = 0; kc < DIN; kc += KC) {
        // stage W chunk pre-swizzled into K-pair layout (coalesced float4 reads)
        for (int i = tid; i < (KC / 2) * (F / 4); i += 128) {   // 32*16 = 512 quads
            int r2 = i / (F / 4);
            int c4 = (i % (F / 4)) * 4;
            const float4 wa = *(const float4*)(W + (kc + 2 * r2    ) * F + c4);
            const float4 wb = *(const float4*)(W + (kc + 2 * r2 + 1) * F + c4);
            sW2[r2][c4 + 0] = (v2f){wa.x, wb.x};
            sW2[r2][c4 + 1] = (v2f){wa.y, wb.y};
            sW2[r2][c4 + 2] = (v2f){wa.z, wb.z};
            sW2[r2][c4 + 3] = (v2f){wa.w, wb.w};
        }
        // stage A chunk (16 x KC) per wave, rows clamped for the tail tile
        for (int i = lane; i < 16 * KC / 4; i += 32) {
            int lin = i * 4, r = lin / KC, c = lin % KC;
            int row = tileRow + r; if (row >= N) row = N - 1;
            float4 x4 = *(const float4*)(X + (long long)row * DIN + kc + c);
            *(float4*)(&sA[wave][r][c]) = x4;
        }
        __syncthreads();

        #pragma unroll
        for (int kk = 0; kk < KC; kk += 4) {
            const int ko = kk + 2 * half;
            const int k2 = ko >> 1;
            v2f a;
            a[0] = sA[wave][l16][ko];
            a[1] = sA[wave][l16][ko + 1];
            v2f b0 = sW2[k2][ 0 + l16];
            v2f b1 = sW2[k2][16 + l16];
            v2f b2 = sW2[k2][32 + l16];
            v2f b3 = sW2[k2][48 + l16];
            acc0 = __builtin_amdgcn_wmma_f32_16x16x4_f32(false, a, false, b0, (short)0, acc0, false, false);
            acc1 = __builtin_amdgcn_wmma_f32_16x16x4_f32(false, a, false, b1, (short)0, acc1, false, false);
            acc2 = __builtin_amdgcn_wmma_f32_16x16x4_f32(false, a, false, b2, (short)0, acc2, false, false);
            acc3 = __builtin_amdgcn_wmma_f32_16x16x4_f32(false, a, false, b3, (short)0, acc3, false, false);
        }
        __syncthreads();
    }

    // D layout: VGPR v, half h -> row M = v + 8h, col N = l16 (+16*tile)
    #pragma unroll
    for (int v = 0; v < 8; v++) {
        int m = tileRow + v + 8 * half;
        if (m < N) {
            float* hrow = H + (long long)m * F;
            hrow[ 0 + l16] = leaky(acc0[v]);
            hrow[16 + l16] = leaky(acc1[v]);
            hrow[32 + l16] = leaky(acc2[v]);
            hrow[48 + l16] = leaky(acc3[v]);
        }
    }
}

// ------------------------------------------------- per-node p/q scalars
__global__ void k_node(const float* __restrict__ Hh, const float* __restrict__ Hl,
                       const float* __restrict__ params,
                       float* __restrict__ ph, float* __restrict__ qh,
                       float* __restrict__ pl, float* __restrict__ ql, int N) {
    int n = blockIdx.x * blockDim.x + threadIdx.x;
    if (n >= N) return;
    const float* hh = Hh + (long long)n * F;
    const float* hl = Hl + (long long)n * F;
    float s0 = 0.f, s1 = 0.f, s2 = 0.f, s3 = 0.f;
    #pragma unroll 8
    for (int j = 0; j < F; j++) {
        float vh = hh[j], vl = hl[j];
        s0 = fmaf(vh, params[0*F + j], s0);
        s1 = fmaf(vh, params[1*F + j], s1);
        s2 = fmaf(vh, params[2*F + j], fmaf(vl, params[4*F + j], s2));
        s3 = fmaf(vh, params[3*F + j], fmaf(vl, params[5*F + j], s3));
    }
    ph[n] = s0; qh[n] = s1; pl[n] = s2; ql[n] = s3;
}

// ------------------------------------------------- edge scores + rowsums
__global__ void k_edge_score(const int* __restrict__ src, const int* __restrict__ dst,
                             const float* __restrict__ ph, const float* __restrict__ qh,
                             const float* __restrict__ pl, const float* __restrict__ ql,
                             const float* __restrict__ params,
                             float* __restrict__ eh, float* __restrict__ el,
                             float* __restrict__ rsh, float* __restrict__ rsl, int E) {
    int e = blockIdx.x * blockDim.x + threadIdx.x;
    if (e >= E) return;
    int s = src[e], d = dst[e];
    float inh = params[6*F + 0], inl = params[6*F + 1];
    float sh = (ph[s] + qh[d]) * inh;
    float sl = (pl[s] + ql[d]) * inl;
    float vh = __expf(-leaky(sh));
    float vl = __expf(-leaky(sl));
    eh[e] = vh; el[e] = vl;
    atomicAdd(rsh + s, vh);
    atomicAdd(rsl + s, vl);
}

// ------------------------------------------------- wave-per-edge aggregation
// lane j handles features j and j+32 of each branch; accumulate into out[src].
__global__ void __launch_bounds__(256)
k_edge_agg(const int* __restrict__ src, const int* __restrict__ dst,
           const float* __restrict__ eh, const float* __restrict__ el,
           const float* __restrict__ Hh, const float* __restrict__ Hl,
           float* __restrict__ out, int E) {
    const int lane = threadIdx.x & 31;
    const int wave = threadIdx.x >> 5;
    int e = blockIdx.x * 8 + wave;
    if (e >= E) return;
    int s = src[e], d = dst[e];
    float vh = eh[e], vl = el[e];
    const float* hh = Hh + (long long)d * F;
    const float* hl = Hl + (long long)d * F;
    float h0 = hh[lane], h1 = hh[32 + lane];
    float l0 = hl[lane], l1 = hl[32 + lane];
    float* o = out + (long long)s * 128;
    atomicAdd(o +       lane, vh * h0);
    atomicAdd(o +  32 + lane, vh * h1);
    atomicAdd(o +  64 + lane, vl * l0);
    atomicAdd(o +  96 + lane, vl * l1);
}

// ------------------------------------------------- normalize + leaky (== relu_bt)
__global__ void k_final(float* __restrict__ out, const float* __restrict__ rsh,
                        const float* __restrict__ rsl, const float* __restrict__ params,
                        int N) {
    long long i = (long long)blockIdx.x * blockDim.x + threadIdx.x;
    long long total = (long long)N * 128;
    if (i >= total) return;
    int n = (int)(i >> 7);
    int j = (int)(i & 127);
    float th = params[6*F + 2], tl = params[6*F + 3];
    float denom = (j < 64) ? (rsh[n] + th) : (rsl[n] + tl);
    out[i] = leaky(out[i] / denom);
}

// ----------------------------------------------------------------------------
extern "C" void kernel_launch(void* const* d_in, const int* in_sizes, int n_in,
                              void* d_out, int out_size, void* d_ws, size_t ws_size,
                              hipStream_t stream) {
    const float* X  = (const float*)d_in[0];
    const int*   ed = (const int*)  d_in[1];
    const float* Wh = (const float*)d_in[2];
    const float* Wl = (const float*)d_in[3];
    const float* ah = (const float*)d_in[4];
    const float* al = (const float*)d_in[5];
    const float* ch = (const float*)d_in[6];
    const float* cl = (const float*)d_in[7];
    float* out = (float*)d_out;

    const int N = in_sizes[0] / DIN;
    const int E = in_sizes[1] / 2;
    const int* src = ed;
    const int* dst = ed + E;

    float* ws = (float*)d_ws;
    size_t off = 0;
    float* Hh  = ws + off; off += (size_t)N * F;
    float* Hl  = ws + off; off += (size_t)N * F;
    float* ph  = ws + off; off += N;
    float* qh  = ws + off; off += N;
    float* pl  = ws + off; off += N;
    float* ql  = ws + off; off += N;
    float* rsh = ws + off; off += N;
    float* rsl = ws + off; off += N;
    float* eh  = ws + off; off += E;
    float* el  = ws + off; off += E;
    float* params = ws + off; off += 6 * F + 8;

    k_zero<<<4096, 256, 0, stream>>>(out, rsh, rsl, N);
    k_params<<<1, 256, 0, stream>>>(ah, al, ch, cl, params);

    const int rowTiles = (N + 15) / 16;
    const int gemmBlocks = (rowTiles + 3) / 4;
    k_gemm_leaky<<<gemmBlocks, 128, 0, stream>>>(X, Wh, Hh, N);
    k_gemm_leaky<<<gemmBlocks, 128, 0, stream>>>(X, Wl, Hl, N);

    k_node<<<(N + 255) / 256, 256, 0, stream>>>(Hh, Hl, params, ph, qh, pl, ql, N);
    k_edge_score<<<(E + 255) / 256, 256, 0, stream>>>(src, dst, ph, qh, pl, ql, params,
                                                      eh, el, rsh, rsl, E);
    k_edge_agg<<<(E + 7) / 8, 256, 0, stream>>>(src, dst, eh, el, Hh, Hl, out, E);
    k_final<<<(int)(((long long)N * 128 + 255) / 256), 256, 0, stream>>>(out, rsh, rsl,
                                                                         params, N);
}